// MultiHeadAttention_18605798326770
// MI455X (gfx1250) — compile-verified
//
#include <hip/hip_runtime.h>

// ---------------------------------------------------------------------------
// Causal MHA forward for MI455X (gfx1250, wave32).
//   Stage 1: qkv = x @ W_attn + b_attn, scattered into Q/K/V [B,H,T,64]
//   Stage 2: flash attention (online softmax), y -> Y [B,T,C]
//   Stage 3: out = Y @ W_proj + b_proj
// fp32 XDL path (V_WMMA_F32_16X16X4_F32) to match the fp32 reference.
// A-panels / K-tiles staged with GLOBAL_LOAD_ASYNC_TO_LDS_B128 (ASYNCcnt);
// B/V staged transposed so every WMMA operand is one contiguous ds_load_b64.
// ---------------------------------------------------------------------------

#define EMB   1024
#define NHEAD 16
#define TSEQ  2048
#define BATCH 2
#define HD    64
#define MROWS (BATCH * TSEQ)   // 4096
#define N_QKV (3 * EMB)        // 3072

typedef float v2f __attribute__((ext_vector_type(2)));
typedef float v8f __attribute__((ext_vector_type(8)));

__device__ __forceinline__ v8f wmma_f32(v2f a, v2f b, v8f c) {
  // (neg_a, A, neg_b, B, c_mod, C, reuse_a, reuse_b)
  return __builtin_amdgcn_wmma_f32_16x16x4_f32(false, a, false, b,
                                               (short)0, c, false, false);
}

// Wave-relative LDS byte offset: low 32 bits of the generic pointer are the
// LDS offset (generic LDS addr = {shared_aperture, 32'offset} per ISA 10.2).
__device__ __forceinline__ unsigned lds_off(const void* p) {
  return (unsigned)(uintptr_t)p;
}

// Async memory->LDS DMA (16B), tracked by ASYNCcnt. vdst = LDS byte address.
__device__ __forceinline__ void async_to_lds_b128(unsigned lds_addr,
                                                  const void* gaddr) {
  asm volatile("global_load_async_to_lds_b128 %0, %1, off"
               :: "v"(lds_addr), "v"(gaddr)
               : "memory");
}

__device__ __forceinline__ void wait_async0() {
  asm volatile("s_wait_asynccnt 0x0" ::: "memory");
}

// ---------------------------------------------------------------------------
// GEMM stage: block = 256 thr (8 waves) -> 64x128 C tile, K panel = 16.
// Wave (wm = w>>1, wn = w&1) owns rows wm*16..+15, cols wn*64..+63
// (4 accumulators -> 16 WMMA per K panel per wave).
// ---------------------------------------------------------------------------
template <int NDIM, int MODE>   // MODE 0: scatter to Q/K/V; MODE 1: plain out
__global__ __launch_bounds__(256) void gemm_kernel(
    const float* __restrict__ X, const float* __restrict__ W,
    const float* __restrict__ bias, float* __restrict__ O0,
    float* __restrict__ O1, float* __restrict__ O2) {
  __shared__ float As[64][20];     // [m][k], rows 80B (16B aligned for async)
  __shared__ float Bst[128][18];   // B transposed: [n][k], rows 72B (8B align)

  const int tid  = threadIdx.x;
  const int lane = tid & 31, l16 = lane & 15, hi = lane >> 4;
  const int w = tid >> 5, wm = w >> 1, wn = w & 1;
  const int m0 = blockIdx.y * 64;
  const int n0 = blockIdx.x * 128;

  v8f acc[4] = {};

  const int la_r = tid >> 2,  la_c = (tid & 3)  << 2;   // A: 64 x 16
  const int lb_r = tid >> 4,  lb_c = (tid & 15) << 3;   // B: 16 x 128, 8 f/thr

  for (int k0 = 0; k0 < EMB; k0 += 16) {
    // A panel: async DMA straight into LDS (contiguous 16B rows).
    async_to_lds_b128(lds_off(&As[la_r][la_c]),
                      &X[(size_t)(m0 + la_r) * EMB + k0 + la_c]);
    // B panel: load 8 floats, store transposed for contiguous fragments.
    {
      const float* Wp = &W[(size_t)(k0 + lb_r) * NDIM + n0 + lb_c];
      float4 b0 = *(const float4*)Wp;
      float4 b1 = *(const float4*)(Wp + 4);
      Bst[lb_c + 0][lb_r] = b0.x; Bst[lb_c + 1][lb_r] = b0.y;
      Bst[lb_c + 2][lb_r] = b0.z; Bst[lb_c + 3][lb_r] = b0.w;
      Bst[lb_c + 4][lb_r] = b1.x; Bst[lb_c + 5][lb_r] = b1.y;
      Bst[lb_c + 6][lb_r] = b1.z; Bst[lb_c + 7][lb_r] = b1.w;
    }
    if (k0 + 16 < EMB) {  // prefetch next panels (global_prefetch_b8)
      __builtin_prefetch(&X[(size_t)(m0 + la_r) * EMB + k0 + 16 + la_c], 0, 0);
      __builtin_prefetch(&W[(size_t)(k0 + 16 + lb_r) * NDIM + n0 + lb_c], 0, 0);
    }
    wait_async0();
    __syncthreads();

    const int arow = wm * 16 + l16;
#pragma unroll
    for (int kk = 0; kk < 4; ++kk) {
      const int kb = kk * 4 + 2 * hi;                    // per-lane K slot
      const v2f a = *(const v2f*)&As[arow][kb];
#pragma unroll
      for (int j = 0; j < 4; ++j) {
        const int col = wn * 64 + j * 16 + l16;
        const v2f b = *(const v2f*)&Bst[col][kb];
        acc[j] = wmma_f32(a, b, acc[j]);
      }
    }
    __syncthreads();
  }

  // Epilogue: bias + store.
#pragma unroll
  for (int j = 0; j < 4; ++j) {
    const int n    = n0 + wn * 64 + j * 16 + l16;
    const float bv = bias[n];
    if (MODE == 0) {   // scatter into Q/K/V [B,H,T,64]
      const int part = n >> 10, c = n & 1023, h = c >> 6, d = c & 63;
      float* base = (part == 0) ? O0 : ((part == 1) ? O1 : O2);
#pragma unroll
      for (int r = 0; r < 8; ++r) {
        const int rowg = m0 + wm * 16 + r + 8 * hi;
        const int bb = rowg >> 11, tt = rowg & (TSEQ - 1);
        base[(((size_t)bb * NHEAD + h) * TSEQ + tt) * HD + d] = acc[j][r] + bv;
      }
    } else {           // row-major out [M, NDIM]
#pragma unroll
      for (int r = 0; r < 8; ++r) {
        const int rowg = m0 + wm * 16 + r + 8 * hi;
        O0[(size_t)rowg * NDIM + n] = acc[j][r] + bv;
      }
    }
  }
}

// ---------------------------------------------------------------------------
// Stage 2: flash attention. Block = 128 thr (4 waves), grid (T/64, B*H).
// Wave w owns q rows q0..q0+15; 16-row KV tiles staged in LDS
// (K via async DMA, V transposed for contiguous PV fragments).
// ---------------------------------------------------------------------------
__global__ __launch_bounds__(128) void attn_kernel(
    const float* __restrict__ Q, const float* __restrict__ K,
    const float* __restrict__ V, float* __restrict__ Y) {
  __shared__ float Ks[16][68];       // [kv][d], rows 272B (16B aligned)
  __shared__ float Vst[64][20];      // V transposed: [d][kv], rows 80B
  __shared__ float Ps[4][16][20];    // per-wave P transpose scratch

  const int tid  = threadIdx.x;
  const int lane = tid & 31, l16 = lane & 15, hi = lane >> 4;
  const int w  = tid >> 5;
  const int bh = blockIdx.y;
  const int q0 = blockIdx.x * 64 + w * 16;

  // Q tile [16 x 64] as 16 A-fragments (K=4 each), kept in registers.
  const float* Qb = Q + ((size_t)bh * TSEQ + q0) * HD;
  v2f aq[16];
#pragma unroll
  for (int kk = 0; kk < 16; ++kk) {
    const int kb = kk * 4 + 2 * hi;
    aq[kk] = *(const v2f*)&Qb[l16 * HD + kb];
  }

  v8f o[4] = {};
  float m_run[8], l_run[8];
#pragma unroll
  for (int r = 0; r < 8; ++r) { m_run[r] = -3.0e38f; l_run[r] = 0.0f; }

  const int ld_r = tid >> 3, ld_c = (tid & 7) << 3;  // 16 x 64, 8 floats/thr
  const int ntiles = blockIdx.x * 4 + 4;             // causal upper bound

  for (int t = 0; t < ntiles; ++t) {
    const int kv0 = t * 16;
    {
      const float* Kp = K + ((size_t)bh * TSEQ + kv0 + ld_r) * HD + ld_c;
      const float* Vp = V + ((size_t)bh * TSEQ + kv0 + ld_r) * HD + ld_c;
      // K: async DMA into LDS (row-major, contiguous 16B chunks).
      async_to_lds_b128(lds_off(&Ks[ld_r][ld_c]),     Kp);
      async_to_lds_b128(lds_off(&Ks[ld_r][ld_c + 4]), Kp + 4);
      // V: load + transposed store -> contiguous PV B-fragments.
      float4 va = *(const float4*)Vp, vb4 = *(const float4*)(Vp + 4);
      Vst[ld_c + 0][ld_r] = va.x;  Vst[ld_c + 1][ld_r] = va.y;
      Vst[ld_c + 2][ld_r] = va.z;  Vst[ld_c + 3][ld_r] = va.w;
      Vst[ld_c + 4][ld_r] = vb4.x; Vst[ld_c + 5][ld_r] = vb4.y;
      Vst[ld_c + 6][ld_r] = vb4.z; Vst[ld_c + 7][ld_r] = vb4.w;
    }
    wait_async0();
    __syncthreads();

    if (kv0 <= q0 + 15) {          // wave-uniform: EXEC stays all-ones
      // S = Q * K^T  (B-frag = contiguous float2 from Ks row n = lane&15)
      v8f s = {};
#pragma unroll
      for (int kk = 0; kk < 16; ++kk) {
        const int kb = kk * 4 + 2 * hi;
        const v2f b = *(const v2f*)&Ks[l16][kb];
        s = wmma_f32(aq[kk], b, s);
      }

      // Online softmax per accumulator row r (global row = q0 + r + 8*hi)
#pragma unroll
      for (int r = 0; r < 8; ++r) {
        const int qrow = q0 + r + 8 * hi;
        const int kcol = kv0 + l16;
        float sv = (kcol <= qrow) ? s[r] * 0.125f : -3.0e38f;
        float rmax = sv;
        rmax = fmaxf(rmax, __shfl_xor(rmax, 1, 32));
        rmax = fmaxf(rmax, __shfl_xor(rmax, 2, 32));
        rmax = fmaxf(rmax, __shfl_xor(rmax, 4, 32));
        rmax = fmaxf(rmax, __shfl_xor(rmax, 8, 32));
        const float mo = m_run[r];
        const float mn = fmaxf(mo, rmax);
        const float p  = __expf(sv - mn);
        float rsum = p;
        rsum += __shfl_xor(rsum, 1, 32);
        rsum += __shfl_xor(rsum, 2, 32);
        rsum += __shfl_xor(rsum, 4, 32);
        rsum += __shfl_xor(rsum, 8, 32);
        const float corr = __expf(mo - mn);
        l_run[r] = l_run[r] * corr + rsum;
        m_run[r] = mn;
#pragma unroll
        for (int nn = 0; nn < 4; ++nn) o[nn][r] *= corr;
        Ps[w][r + 8 * hi][l16] = p;   // C-layout -> LDS
      }

      // O += P * V   (A-frag from Ps, B-frag = contiguous float2 from Vst;
      //  same-wave DS ops are in-order, no barrier needed for the P hop)
#pragma unroll
      for (int kk = 0; kk < 4; ++kk) {
        const int kb = kk * 4 + 2 * hi;
        const v2f a = *(const v2f*)&Ps[w][l16][kb];
#pragma unroll
        for (int nn = 0; nn < 4; ++nn) {
          const v2f b = *(const v2f*)&Vst[nn * 16 + l16][kb];
          o[nn] = wmma_f32(a, b, o[nn]);
        }
      }
    }
    __syncthreads();
  }

  // Normalize and write y -> Y [B,T,C]
  const int b = bh >> 4, h = bh & 15;
#pragma unroll
  for (int r = 0; r < 8; ++r) {
    const float inv = 1.0f / l_run[r];
    const int gq = q0 + r + 8 * hi;
    float* yrow = Y + ((size_t)b * TSEQ + gq) * EMB + h * HD;
#pragma unroll
    for (int nn = 0; nn < 4; ++nn) yrow[nn * 16 + l16] = o[nn][r] * inv;
  }
}

// ---------------------------------------------------------------------------
extern "C" void kernel_launch(void* const* d_in, const int* in_sizes, int n_in,
                              void* d_out, int out_size, void* d_ws,
                              size_t ws_size, hipStream_t stream) {
  const float* x      = (const float*)d_in[0];
  const float* W_attn = (const float*)d_in[1];
  const float* b_attn = (const float*)d_in[2];
  const float* W_proj = (const float*)d_in[3];
  const float* b_proj = (const float*)d_in[4];
  float* out = (float*)d_out;

  // Workspace: Q, K, V [B,H,T,64] + Y [B,T,C]  (4 x 16.8 MB = 67 MB)
  const size_t plane = (size_t)MROWS * EMB;   // 4096*1024 floats
  float* Q = (float*)d_ws;
  float* K = Q + plane;
  float* V = K + plane;
  float* Y = V + plane;

  gemm_kernel<N_QKV, 0><<<dim3(N_QKV / 128, MROWS / 64), 256, 0, stream>>>(
      x, W_attn, b_attn, Q, K, V);
  attn_kernel<<<dim3(TSEQ / 64, BATCH * NHEAD), 128, 0, stream>>>(Q, K, V, Y);
  gemm_kernel<EMB, 1><<<dim3(EMB / 128, MROWS / 64), 256, 0, stream>>>(
      Y, W_proj, b_proj, out, nullptr, nullptr);
}